// AttentivePooling_64046552318008
// MI455X (gfx1250) — compile-verified
//
#include <hip/hip_runtime.h>
#include <hip/hip_bf16.h>
#include <math.h>

// CDNA5 (gfx1250) wave32 WMMA types
typedef float v2f __attribute__((ext_vector_type(2)));
typedef float v8f __attribute__((ext_vector_type(8)));

#define DIMK 256      // DIM
#define ODIM 256      // OUT_DIM
#define NSEG 8        // K-split segments for the pooling GEMM
#define SEGK 6400     // ceil(50000/8) rounded up to a multiple of 256

// Async global->LDS DMA (ASYNCcnt-tracked). ldsoff = workgroup-relative LDS
// byte offset (low 32 bits of the generic pointer), gp = global address.
__device__ __forceinline__ void async_copy16(unsigned ldsoff, const float* gp) {
    asm volatile("global_load_async_to_lds_b128 %0, %1, off"
                 :: "v"(ldsoff), "v"(gp) : "memory");
}
__device__ __forceinline__ void async_copy4(unsigned ldsoff, const float* gp) {
    asm volatile("global_load_async_to_lds_b32 %0, %1, off"
                 :: "v"(ldsoff), "v"(gp) : "memory");
}
__device__ __forceinline__ void async_wait0() {
    asm volatile("s_wait_asynccnt 0x0" ::: "memory");
}

// ---------------------------------------------------------------------------
// Kernel 1: h = leaky_relu(X @ W + b), X:[N,256], W:[256,256], h:[N,256]
// 8 waves/block, each wave owns a 16-row strip. W panel of 2 column tiles
// staged TRANSPOSED in LDS (sW[cc][k], 32x256 f32 = 32KB) via per-element
// async B32 DMA scatter, so each B fragment {B[k],B[k+1]} is a contiguous
// float2 -> single ds_load_b64 straight into the WMMA operand pair (no movs).
// Each A fragment load feeds 2 f32 WMMAs.
// ---------------------------------------------------------------------------
__global__ __launch_bounds__(256) void k_h_gemm(const float* __restrict__ X,
                                                const float* __restrict__ W,
                                                const float* __restrict__ bias,
                                                float* __restrict__ H,
                                                int nrows) {
    __shared__ float sW[32 * DIMK];   // [cc][k] (transposed panel)

    const int tid  = threadIdx.x;
    const int wave = tid >> 5;
    const int lane = tid & 31;
    const int half = lane >> 4;       // 0: lanes 0-15, 1: lanes 16-31
    const int l16  = lane & 15;
    const int row0 = blockIdx.x * 128 + wave * 16;

    // Clamp A-row for loads (WMMA needs EXEC all-1s; stores are guarded)
    int arow = row0 + l16;
    if (arow >= nrows) arow = nrows - 1;
    const float* __restrict__ xrow = X + (size_t)arow * DIMK;

    for (int jp = 0; jp < ODIM / 32; ++jp) {          // 8 panels of 32 cols
        // Async-stage W[:, jp*32 : jp*32+32] transposed into LDS.
        // Global side coalesced (consecutive lanes -> consecutive cols),
        // LDS side scattered (transpose) via independent per-op dest addrs.
        for (int g = tid; g < 32 * DIMK; g += 256) {
            int kk = g >> 5;             // k index
            int cc = g & 31;             // panel-local column
            unsigned lds = (unsigned)(size_t)&sW[cc * DIMK + kk];
            async_copy4(lds, W + (size_t)kk * ODIM + jp * 32 + cc);
        }
        async_wait0();
        __syncthreads();

        // C init = bias (C layout: col = lane%16 in every C VGPR)
        v8f acc0, acc1;
        float b0v = bias[jp * 32 + l16];
        float b1v = bias[jp * 32 + 16 + l16];
#pragma unroll
        for (int i = 0; i < 8; ++i) { acc0[i] = b0v; acc1[i] = b1v; }

#pragma unroll 4
        for (int k = 0; k < DIMK; k += 4) {
            v2f a, b0, b1;
            // A frag: lanes 0-15 -> (K0,K1), lanes 16-31 -> (K2,K3) of row l16
            a[0] = xrow[k + 2 * half + 0];
            a[1] = xrow[k + 2 * half + 1];
            // B frags: contiguous {K,K+1} per column -> ds_load_b64
            const float2 bv0 = *(const float2*)&sW[l16 * DIMK + k + 2 * half];
            const float2 bv1 = *(const float2*)&sW[(16 + l16) * DIMK + k + 2 * half];
            b0[0] = bv0.x; b0[1] = bv0.y;
            b1[0] = bv1.x; b1[1] = bv1.y;
            acc0 = __builtin_amdgcn_wmma_f32_16x16x4_f32(false, a, false, b0,
                                                         (short)0, acc0, false, false);
            acc1 = __builtin_amdgcn_wmma_f32_16x16x4_f32(false, a, false, b1,
                                                         (short)0, acc1, false, false);
        }

        // Leaky ReLU + store (C VGPR i: row = i (lo half) / i+8 (hi half))
#pragma unroll
        for (int i = 0; i < 8; ++i) {
            int row = row0 + i + 8 * half;
            if (row < nrows) {
                float v0 = acc0[i];
                float v1 = acc1[i];
                v0 = (v0 > 0.0f) ? v0 : 0.01f * v0;
                v1 = (v1 > 0.0f) ? v1 : 0.01f * v1;
                H[(size_t)row * ODIM + jp * 32 + l16]      = v0;
                H[(size_t)row * ODIM + jp * 32 + 16 + l16] = v1;
            }
        }
        __syncthreads();
    }
}

// ---------------------------------------------------------------------------
// Kernel 2: a[n] = X[n,:] @ w_align + b_align  (memory-bound GEMV)
// ---------------------------------------------------------------------------
__global__ __launch_bounds__(256) void k_align(const float* __restrict__ X,
                                               const float* __restrict__ w,
                                               const float* __restrict__ b,
                                               float* __restrict__ A,
                                               int nrows) {
    int row = blockIdx.x * blockDim.x + threadIdx.x;
    if (row >= nrows) return;
    const float* __restrict__ xr = X + (size_t)row * DIMK;
    float s = 0.0f;
#pragma unroll 8
    for (int k = 0; k < DIMK; ++k) s = fmaf(xr[k], w[k], s);
    A[row] = s + b[0];
}

// ---------------------------------------------------------------------------
// Kernel 3: row softmax of logits = mnm*a + mask, written into Wt [M,N].
// One block per molecule row; online (max,sum) streaming pass storing raw
// logits into Wt, LDS pair-reduction, then in-place normalize pass.
// ---------------------------------------------------------------------------
__global__ __launch_bounds__(256) void k_softmax(const float* __restrict__ mnm,
                                                 const float* __restrict__ mask,
                                                 const float* __restrict__ A,
                                                 float* __restrict__ Wt,
                                                 int Ncols) {
    __shared__ float smx[256];
    __shared__ float ssm[256];
    const int m   = blockIdx.x;
    const int tid = threadIdx.x;
    const size_t base = (size_t)m * Ncols;

    float mx = -3.4e38f;
    float sm = 0.0f;
    for (int n = tid; n < Ncols; n += 256) {
        float l = fmaf(mnm[base + n], A[n], mask[base + n]);
        Wt[base + n] = l;                      // stash raw logit
        float nm = fmaxf(mx, l);
        sm = sm * __expf(mx - nm) + __expf(l - nm);
        mx = nm;
    }
    smx[tid] = mx;
    ssm[tid] = sm;
    __syncthreads();
    for (int off = 128; off > 0; off >>= 1) {
        if (tid < off) {
            float m2 = smx[tid + off], s2 = ssm[tid + off];
            float nm = fmaxf(smx[tid], m2);
            ssm[tid] = ssm[tid] * __expf(smx[tid] - nm) + s2 * __expf(m2 - nm);
            smx[tid] = nm;
        }
        __syncthreads();
    }
    const float Mx  = smx[0];
    const float inv = 1.0f / ssm[0];
    for (int n = tid; n < Ncols; n += 256) {
        Wt[base + n] = __expf(Wt[base + n] - Mx) * inv;
    }
}

// ---------------------------------------------------------------------------
// Kernel 4: partial GEMM  P[seg] = Wt[:, seg-cols] @ H[seg-rows, :]
// Grid: (M/16 row tiles * 2 col halves, NSEG) = 512 blocks -> enough WGPs to
// saturate the 205MB weight stream. Wt K-chunk (16x256 = 16KB) async-B128
// DMA'd into LDS; B frags stream from H (51MB, L2-resident).
// ---------------------------------------------------------------------------
__global__ __launch_bounds__(256) void k_pool_gemm(const float* __restrict__ Wt,
                                                   const float* __restrict__ H,
                                                   float* __restrict__ P,
                                                   int Ncols) {
    __shared__ float sA[16 * 256];    // [row][k], row stride 256 floats

    const int tid  = threadIdx.x;
    const int wave = tid >> 5;
    const int lane = tid & 31;
    const int half = lane >> 4;
    const int l16  = lane & 15;
    const int r0   = ((int)blockIdx.x >> 1) * 16;
    const int c0   = ((int)blockIdx.x & 1) * 128 + wave * 16;
    const int seg  = blockIdx.y;
    const int ks   = seg * SEGK;
    const int ke   = min(Ncols, ks + SEGK);

    v8f acc;
#pragma unroll
    for (int i = 0; i < 8; ++i) acc[i] = 0.0f;

    for (int kc = ks; kc < ke; kc += 256) {
        const int len  = min(256, ke - kc);  // 256 or 80; both %4 == 0
        const int chpr = len >> 2;           // 16B chunks per row
        __syncthreads();
        // Async-stage Wt[r0:r0+16, kc:kc+len] into LDS (16B DMA chunks).
        for (int c = tid; c < 16 * chpr; c += 256) {
            int rr = c / chpr;
            int k4 = (c - rr * chpr) * 4;
            unsigned lds = (unsigned)(size_t)&sA[rr * 256 + k4];
            async_copy16(lds, Wt + (size_t)(r0 + rr) * Ncols + kc + k4);
        }
        async_wait0();
        __syncthreads();

        for (int kk = 0; kk < len; kk += 4) {
            v2f a, b;
            // A frag from LDS (8B-aligned float2 -> ds_load_b64)
            const float2 av = *(const float2*)&sA[l16 * 256 + kk + 2 * half];
            a[0] = av.x;
            a[1] = av.y;
            // B frag from H: rows (kc+kk+2*half)+{0,1}, col c0+l16
            const float* __restrict__ hp =
                H + (size_t)(kc + kk + 2 * half) * ODIM + c0 + l16;
            __builtin_prefetch(hp + 8 * ODIM, 0, 0);  // global_prefetch ahead of stream
            b[0] = hp[0];
            b[1] = hp[ODIM];
            acc = __builtin_amdgcn_wmma_f32_16x16x4_f32(false, a, false, b,
                                                        (short)0, acc, false, false);
        }
    }

    float* __restrict__ Pseg = P + (size_t)seg * 512 * ODIM;
#pragma unroll
    for (int i = 0; i < 8; ++i) {
        Pseg[(size_t)(r0 + i + 8 * half) * ODIM + c0 + l16] = acc[i];
    }
}

// ---------------------------------------------------------------------------
// Kernel 5: Out[i] = sum over NSEG partials (fixed order -> deterministic)
// ---------------------------------------------------------------------------
__global__ __launch_bounds__(256) void k_reduce(const float* __restrict__ P,
                                                float* __restrict__ Out,
                                                int total) {
    int i = blockIdx.x * blockDim.x + threadIdx.x;
    if (i >= total) return;
    float s = 0.0f;
#pragma unroll
    for (int sgi = 0; sgi < NSEG; ++sgi) s += P[(size_t)sgi * total + i];
    Out[i] = s;
}

// ---------------------------------------------------------------------------
// Launcher
// ---------------------------------------------------------------------------
extern "C" void kernel_launch(void* const* d_in, const int* in_sizes, int n_in,
                              void* d_out, int out_size, void* d_ws, size_t ws_size,
                              hipStream_t stream) {
    const float* X    = (const float*)d_in[0];   // [N, 256]
    const float* MNM  = (const float*)d_in[1];   // [M, N]
    const float* MASK = (const float*)d_in[2];   // [M, N]
    const float* Wat  = (const float*)d_in[3];   // [256, 256]
    const float* Bat  = (const float*)d_in[4];   // [256]
    const float* Wal  = (const float*)d_in[5];   // [256]
    const float* Bal  = (const float*)d_in[6];   // [1]

    const int N = in_sizes[0] / DIMK;            // 50000
    const int M = in_sizes[1] / N;               // 512

    float* Out = (float*)d_out;                    // [M, 256]  (tuple elem 0)
    float* Wt  = (float*)d_out + (size_t)M * ODIM; // [M, N]    (tuple elem 1)

    float* H = (float*)d_ws;                     // [N, 256] scratch
    float* A = H + (size_t)N * ODIM;             // [N] scratch
    float* P = A + N;                            // [NSEG, M, 256] partials

    k_h_gemm  <<<(N + 127) / 128, 256, 0, stream>>>(X, Wat, Bat, H, N);
    k_align   <<<(N + 255) / 256, 256, 0, stream>>>(X, Wal, Bal, A, N);
    k_softmax <<<M,               256, 0, stream>>>(MNM, MASK, A, Wt, N);
    k_pool_gemm<<<dim3((M / 16) * 2, NSEG), 256, 0, stream>>>(Wt, H, P, N);
    k_reduce  <<<(M * ODIM + 255) / 256, 256, 0, stream>>>(P, Out, M * ODIM);
}